// MedianBlur_22771916604110
// MI455X (gfx1250) — compile-verified
//
#include <hip/hip_runtime.h>
#include <hip/hip_bf16.h>

// Median blur 3x3 + residual: out = x + 0.2*(med3x3(x) - x)
// x: [8, 64, 256, 256] f32  -> 512 planes of 256x256.
// Bandwidth-bound (256 MiB @ 23.3 TB/s ~ 11.5 us floor). Strategy:
//  - stage 34-row bands into LDS once via CDNA5 async global->LDS DMA (ASYNCcnt)
//  - exact 3x3 median via sorted-column-triples identity using v_med3/min3/max3
//  - coalesced nontemporal f32 stores.

#define IMG_W 256
#define IMG_H 256
#define BAND_H 32
#define LDS_ROWS (BAND_H + 2)

__device__ __forceinline__ float min3f(float a, float b, float c) {
    return __builtin_fminf(__builtin_fminf(a, b), c);
}
__device__ __forceinline__ float max3f(float a, float b, float c) {
    return __builtin_fmaxf(__builtin_fmaxf(a, b), c);
}
__device__ __forceinline__ float med3f(float a, float b, float c) {
    return __builtin_amdgcn_fmed3f(a, b, c);
}

__global__ __launch_bounds__(256) void median3x3_res_kernel(
        const float* __restrict__ x, float* __restrict__ out) {
    __shared__ __align__(16) float tile[LDS_ROWS * IMG_W];

    const int tid  = threadIdx.x;                 // 0..255
    const int bid  = blockIdx.x;
    const int bandsPerPlane = IMG_H / BAND_H;     // 8
    const int p    = bid / bandsPerPlane;         // plane (b*c)
    const int band = bid % bandsPerPlane;
    const int row0 = band * BAND_H;               // first output row of band

    // -------- stage LDS_ROWS rows (global rows row0-1 .. row0+BAND_H) --------
    // Each lane moves 16 B; one async instruction moves 4 rows (4 KB) per block.
    const unsigned ldsBase = (unsigned)(size_t)(&tile[0]); // low 32 bits = LDS byte offset
    const int      jr      = tid >> 6;     // 0..3  : row within 4-row group
    const int      chunk   = tid & 63;     // 0..63 : 16-byte chunk within 1 KB row
    const unsigned planeByteOff = (unsigned)p * (unsigned)(IMG_H * IMG_W * 4);

    for (int j0 = 0; j0 < LDS_ROWS; j0 += 4) {
        const int lrow = j0 + jr;
        if (lrow < LDS_ROWS) {
            const int grow = row0 - 1 + lrow;
            const unsigned ldsOff = ldsBase + (unsigned)(lrow * (IMG_W * 4) + chunk * 16);
            if (grow >= 0 && grow < IMG_H) {
                const unsigned gOff =
                    planeByteOff + (unsigned)(grow * (IMG_W * 4) + chunk * 16);
                // CDNA5 async DMA: copies 16 B/lane from global address x + gOff to LDS address ldsOff,
                // tracked by ASYNCcnt.
                asm volatile("global_load_async_to_lds_b128 %0, %1, %2"
                             :: "v"(ldsOff), "v"(gOff), "s"(x)
                             : "memory");
            } else {
                // zero padding rows (matches reference zero pad)
                float4 z = make_float4(0.f, 0.f, 0.f, 0.f);
                *reinterpret_cast<float4*>(&tile[lrow * IMG_W + chunk * 4]) = z;
            }
        }
    }
    asm volatile("s_wait_asynccnt 0" ::: "memory"); // our wave's DMAs done in LDS
    __syncthreads();                                 // all waves' fills visible

    // -------- compute: each thread owns column `cx`, walks BAND_H rows --------
    const int  cx   = tid;
    const bool hasL = (cx > 0);
    const bool hasR = (cx < IMG_W - 1);
    const int  xm   = hasL ? cx - 1 : cx;
    const int  xp   = hasR ? cx + 1 : cx;

    auto ld3 = [&](int lr, float& l, float& c, float& r) {
        const float* rowp = &tile[lr * IMG_W];
        float lv = rowp[xm];
        float rv = rowp[xp];
        c = rowp[cx];
        l = hasL ? lv : 0.0f;   // zero pad left/right edges
        r = hasR ? rv : 0.0f;
    };

    float lf0, ct0, rt0, lf1, ct1, rt1;
    ld3(0, lf0, ct0, rt0);
    ld3(1, lf1, ct1, rt1);

    const size_t outBase = (size_t)p * (IMG_H * IMG_W) + (size_t)row0 * IMG_W + cx;

    #pragma unroll 4
    for (int i = 0; i < BAND_H; ++i) {
        float lf2, ct2, rt2;
        ld3(i + 2, lf2, ct2, rt2);

        // per-column sorted triples (lo <= mid <= hi)
        const float Llo = min3f(lf0, lf1, lf2);
        const float Lmd = med3f(lf0, lf1, lf2);
        const float Lhi = max3f(lf0, lf1, lf2);
        const float Clo = min3f(ct0, ct1, ct2);
        const float Cmd = med3f(ct0, ct1, ct2);
        const float Chi = max3f(ct0, ct1, ct2);
        const float Rlo = min3f(rt0, rt1, rt2);
        const float Rmd = med3f(rt0, rt1, rt2);
        const float Rhi = max3f(rt0, rt1, rt2);

        // exact 3x3 median identity
        const float mlo = max3f(Llo, Clo, Rlo);
        const float mmd = med3f(Lmd, Cmd, Rmd);
        const float mhi = min3f(Lhi, Chi, Rhi);
        const float med = med3f(mlo, mmd, mhi);

        const float xin = ct1;                       // center pixel
        const float res = xin + 0.2f * (med - xin);  // mul+add like reference
        __builtin_nontemporal_store(res, &out[outBase + (size_t)i * IMG_W]);

        lf0 = lf1; lf1 = lf2;
        ct0 = ct1; ct1 = ct2;
        rt0 = rt1; rt1 = rt2;
    }
}

extern "C" void kernel_launch(void* const* d_in, const int* in_sizes, int n_in,
                              void* d_out, int out_size, void* d_ws, size_t ws_size,
                              hipStream_t stream) {
    const float* x   = (const float*)d_in[0];
    float*       out = (float*)d_out;
    const int planes = in_sizes[0] / (IMG_H * IMG_W); // 8*64 = 512
    const dim3 grid(planes * (IMG_H / BAND_H));       // 4096 blocks
    median3x3_res_kernel<<<grid, 256, 0, stream>>>(x, out);
}